// Attention_33560874451088
// MI455X (gfx1250) — compile-verified
//
#include <hip/hip_runtime.h>
#include <stdint.h>

typedef float v2f __attribute__((ext_vector_type(2)));
typedef float v8f __attribute__((ext_vector_type(8)));

#define NUM_SEQS 32
#define NUM_HEADS 32
#define NUM_KV 8
#define GQA 4
#define HD 128
#define BS 16
#define MAXB 128
#define WAVES 4
#define SPLITS 4
#define KSTR 132   /* padded LDS row stride (floats): bank = 4*row+col -> conflict-free */
#define WSSTR (HD + 2)          /* per-(split,seq,kvh,row) ws entry: O[128], M, L */
#define NEG_BIG (-3.0e38f)
#define ATTN_SCALE 0.08838834764831845f

__device__ __forceinline__ v8f wmma_f32(v2f a, v2f b, v8f c) {
  // D = A(16x4 f32) * B(4x16 f32) + C(16x16 f32)
  return __builtin_amdgcn_wmma_f32_16x16x4_f32(false, a, false, b, (short)0, c,
                                               false, false);
}

// Generic pointers to LDS on AMDGPU are {shared-aperture-hi, 32-bit LDS offset};
// the low 32 bits are the wave-relative LDS byte offset the async ops want.
__device__ __forceinline__ uint32_t lds_off(const void* p) {
  return (uint32_t)(uintptr_t)p;
}

__global__ __launch_bounds__(WAVES * 32) void paged_attn_kernel(
    const float* __restrict__ q,
    const float* __restrict__ knew_g,
    const float* __restrict__ vnew_g,
    const float* __restrict__ kcache,
    const float* __restrict__ vcache,
    const int* __restrict__ block_tables,
    const int* __restrict__ ctx_lens,
    float* __restrict__ ws,          // split partials (unused when gridDim.z == 1)
    float* __restrict__ out)         // direct output (used when gridDim.z == 1)
{
  const int kvh   = blockIdx.x;          // 0..7
  const int seq   = blockIdx.y;          // 0..31
  const int split = blockIdx.z;
  const int nspl  = gridDim.z;
  const int wave  = threadIdx.x >> 5;    // 0..3
  const int lane  = threadIdx.x & 31;
  const int ln16  = lane & 15;
  const int hi    = lane >> 4;           // half-wave: A/B frag K-cols {0,1} vs {2,3}

  const int ctx    = ctx_lens[seq];
  const int newest = ctx - 1;            // newest token: substituted from k/v inputs
  const int ntiles = (ctx + BS - 1) / BS;

  __shared__ __align__(16) float q_lds[GQA * KSTR];       // Q tile (block-shared)
  __shared__ __align__(16) float k_lds[WAVES][BS * KSTR]; // per-wave async K tile
  __shared__ float p_lds[WAVES][GQA * BS];                // per-wave P (C->A layout)
  __shared__ float o_lds[WAVES][GQA][HD];                 // per-wave partial O
  __shared__ float m_lds[WAVES][GQA];
  __shared__ float l_lds[WAVES][GQA];

  // ---- cooperative Q stage (4 rows x 128, padded) ----
  for (int i = threadIdx.x; i < GQA * HD; i += WAVES * 32) {
    const int r = i >> 7, c0 = i & (HD - 1);
    q_lds[r * KSTR + c0] =
        q[((size_t)seq * NUM_HEADS + kvh * GQA + r) * HD + c0];
  }
  __syncthreads();

  const float* knew = knew_g + ((size_t)seq * NUM_KV + kvh) * HD;
  const float* vnew = vnew_g + ((size_t)seq * NUM_KV + kvh) * HD;

  v8f acc[HD / 16];
#pragma unroll
  for (int t = 0; t < HD / 16; ++t)
#pragma unroll
    for (int e = 0; e < 8; ++e) acc[t][e] = 0.f;

  float m_i[GQA], l_i[GQA];
#pragma unroll
  for (int r = 0; r < GQA; ++r) { m_i[r] = NEG_BIG; l_i[r] = 0.f; }

  const size_t tokstride = (size_t)NUM_KV * HD;   // 1024 floats between cached tokens

  // A-frag: row (ln16&3) of Q (A rows 4..15 duplicate rows 0..3 -> finite everywhere)
  const float* qf = &q_lds[(ln16 & (GQA - 1)) * KSTR + 2 * hi];
  // B-frag: column n = token ln16 of the staged K tile
  const float* kf = &k_lds[wave][ln16 * KSTR + 2 * hi];
  const uint32_t kdst0 = lds_off(&k_lds[wave][0]) + (uint32_t)(lane * 16);

  for (int tile = split * WAVES + wave; tile < ntiles; tile += nspl * WAVES) {
    const int tb  = tile * BS;
    const int blk = block_tables[seq * MAXB + tile];
    const size_t base = (((size_t)blk * BS) * NUM_KV + kvh) * (size_t)HD;

    // ---- async-stage K tile: 16 coalesced 512B rows -> LDS (ASYNCcnt path).
    //      Uniform base; newest row patched below (no per-row selects). ----
    const uint64_t gbase =
        (uint64_t)(uintptr_t)(kcache + base) + (uint64_t)(lane * 16u);
#pragma unroll
    for (int i = 0; i < BS; ++i) {
      const uint64_t ga = gbase + (uint64_t)(i * 4096u);   // token stride = 4KB
      const uint32_t la = kdst0 + (uint32_t)(i * (KSTR * 4));
      asm volatile("global_load_async_to_lds_b128 %0, %1, off"
                   :: "v"(la), "v"(ga) : "memory");
    }
    asm volatile("s_wait_asynccnt 0x0" ::: "memory");

    if (tb <= newest && newest < tb + BS) {      // wave-uniform branch
      const float4 kv4 = ((const float4*)knew)[lane];          // 16B per lane
      float4* dst = (float4*)&k_lds[wave][(newest - tb) * KSTR + lane * 4];
      *dst = kv4;                                              // overwrite 512B row
    }

    // ---- S = Q K^T over d-chunks of 4 (A,B frags from LDS) ----
    v8f s;
#pragma unroll
    for (int e = 0; e < 8; ++e) s[e] = 0.f;
#pragma unroll
    for (int c = 0; c < HD / 4; ++c) {
      v2f a; a.x = qf[4 * c]; a.y = qf[4 * c + 1];
      v2f b; b.x = kf[4 * c]; b.y = kf[4 * c + 1];
      s = wmma_f32(a, b, s);
    }

    const int tok = tb + ln16;
    const bool valid = tok < ctx;
#pragma unroll
    for (int e = 0; e < 8; ++e) s[e] = valid ? s[e] * ATTN_SCALE : NEG_BIG;

    // ---- online softmax, rows 0..3 (row r in s[r], lanes 0..15) ----
#pragma unroll
    for (int r = 0; r < GQA; ++r) {
      float x = s[r];
      float mx = x;
#pragma unroll
      for (int o = 1; o < 16; o <<= 1) mx = fmaxf(mx, __shfl_xor(mx, o, 32));
      const float mnew  = fmaxf(m_i[r], mx);
      const float alpha = __expf(m_i[r] - mnew);
      const float pv    = valid ? __expf(x - mnew) : 0.f;
      float sum = pv;
#pragma unroll
      for (int o = 1; o < 16; o <<= 1) sum += __shfl_xor(sum, o, 32);
      l_i[r] = l_i[r] * alpha + sum;
      m_i[r] = mnew;
#pragma unroll
      for (int t = 0; t < HD / 16; ++t) acc[t][r] *= alpha;
      if (hi == 0) p_lds[wave][r * BS + ln16] = pv;   // wave-private, in-order DS
    }

    // ---- O += P V : A = P rows (dup 0..3) from LDS, B = V direct from global ----
#pragma unroll
    for (int c = 0; c < 4; ++c) {
      const float* pp = &p_lds[wave][(ln16 & (GQA - 1)) * BS + 4 * c + 2 * hi];
      v2f ap; ap.x = pp[0]; ap.y = pp[1];
      const int tA = 4 * c + 2 * hi;   // token-in-tile for this half-wave's B rows
      const float* vrowA = (tb + tA == newest)
                               ? vnew : (vcache + base + (size_t)tA * tokstride);
      const float* vrowB = (tb + tA + 1 == newest)
                               ? vnew : (vcache + base + (size_t)(tA + 1) * tokstride);
#pragma unroll
      for (int t = 0; t < HD / 16; ++t) {
        v2f bv; bv.x = vrowA[t * 16 + ln16]; bv.y = vrowB[t * 16 + ln16];
        acc[t] = wmma_f32(ap, bv, acc[t]);
      }
    }
  }

  // ---- stash per-wave partials ----
  if (hi == 0) {
#pragma unroll
    for (int r = 0; r < GQA; ++r) {
#pragma unroll
      for (int t = 0; t < HD / 16; ++t)
        o_lds[wave][r][t * 16 + ln16] = acc[t][r];
      if (ln16 == 0) { m_lds[wave][r] = m_i[r]; l_lds[wave][r] = l_i[r]; }
    }
  }
  __syncthreads();

  // ---- merge 4 wave-partials; write output (1 split) or ws partial ----
  const int d = threadIdx.x;   // 0..127
#pragma unroll
  for (int r = 0; r < GQA; ++r) {
    float M = NEG_BIG;
#pragma unroll
    for (int w = 0; w < WAVES; ++w) M = fmaxf(M, m_lds[w][r]);
    float L = 0.f, O = 0.f;
#pragma unroll
    for (int w = 0; w < WAVES; ++w) {
      const float sc = __expf(m_lds[w][r] - M);
      L += l_lds[w][r] * sc;
      O += o_lds[w][r][d] * sc;
    }
    if (nspl == 1) {
      out[(((size_t)seq * NUM_HEADS) + kvh * GQA + r) * HD + d] = O / L;
    } else {
      float* wsrow = ws + ((((size_t)split * NUM_SEQS + seq) * NUM_KV + kvh) * GQA + r)
                              * (size_t)WSSTR;
      wsrow[d] = O;
      if (d == 0) { wsrow[HD] = M; wsrow[HD + 1] = L; }
    }
  }
}

__global__ __launch_bounds__(128) void paged_attn_reduce(
    const float* __restrict__ ws, float* __restrict__ out, int nsplits)
{
  const int kvh = blockIdx.x;
  const int seq = blockIdx.y;
  const int d   = threadIdx.x;  // 0..127
#pragma unroll
  for (int r = 0; r < GQA; ++r) {
    const size_t eidx0 = (((size_t)seq * NUM_KV + kvh) * GQA + r) * (size_t)WSSTR;
    const size_t sstr  = (size_t)NUM_SEQS * NUM_KV * GQA * WSSTR;
    float M = NEG_BIG;
    for (int s = 0; s < nsplits; ++s) M = fmaxf(M, ws[eidx0 + s * sstr + HD]);
    float L = 0.f, O = 0.f;
    for (int s = 0; s < nsplits; ++s) {
      const float* wsrow = ws + eidx0 + s * sstr;
      const float sc = __expf(wsrow[HD] - M);
      L += wsrow[HD + 1] * sc;
      O += wsrow[d] * sc;
    }
    out[(((size_t)seq * NUM_HEADS) + kvh * GQA + r) * HD + d] = O / L;
  }
}

extern "C" void kernel_launch(void* const* d_in, const int* in_sizes, int n_in,
                              void* d_out, int out_size, void* d_ws, size_t ws_size,
                              hipStream_t stream) {
  (void)in_sizes; (void)n_in; (void)out_size;
  const float* q  = (const float*)d_in[0];
  const float* k  = (const float*)d_in[1];
  const float* v  = (const float*)d_in[2];
  const float* kc = (const float*)d_in[3];
  const float* vc = (const float*)d_in[4];
  // d_in[5] = slot_mapping: unused — newest-token slot derived from context_lens,
  // substituted at read time so the input caches are never mutated.
  const int* bt = (const int*)d_in[6];
  const int* cl = (const int*)d_in[7];
  float* out = (float*)d_out;

  const size_t need =
      (size_t)SPLITS * NUM_SEQS * NUM_KV * GQA * WSSTR * sizeof(float);
  if (d_ws != nullptr && ws_size >= need) {
    // Flash-decoding: 4 context splits -> 4096 waves streaming KV concurrently,
    // then a log-sum-exp merge pass.
    paged_attn_kernel<<<dim3(NUM_KV, NUM_SEQS, SPLITS), WAVES * 32, 0, stream>>>(
        q, k, v, kc, vc, bt, cl, (float*)d_ws, out);
    paged_attn_reduce<<<dim3(NUM_KV, NUM_SEQS), 128, 0, stream>>>(
        (const float*)d_ws, out, SPLITS);
  } else {
    // Fallback: single-pass, block-internal merge only.
    paged_attn_kernel<<<dim3(NUM_KV, NUM_SEQS, 1), WAVES * 32, 0, stream>>>(
        q, k, v, kc, vc, bt, cl, nullptr, out);
  }
}